// RADAM_74646531605154
// MI455X (gfx1250) — compile-verified
//
#include <hip/hip_runtime.h>
#include <hip/hip_bf16.h>
#include <math.h>

// Problem constants (from reference): B=128, Z=1024, H=W=28, N=784, M(RAEs)=4
#define BZ   1024
#define BN   784
#define BH   28
#define BW   28
#define BM   4
#define DHALF 512   // Z/2
#define GSTR 800    // padded stride for g rows

typedef float v2f __attribute__((ext_vector_type(2)));
typedef float v8f __attribute__((ext_vector_type(8)));

// div[k] = exp(-2k * ln(1e4)/512)
#define DIV_COEF (-0.03597889073f)

__global__ __launch_bounds__(256)
void rae_fused_kernel(const float* __restrict__ x,
                      const float* __restrict__ Wm,
                      float* __restrict__ out)
{
    const int b    = blockIdx.x;
    const int tid  = threadIdx.x;
    const int wave = tid >> 5;
    const int lane = tid & 31;

    // Row 4 (index BM) of s_a / s_g is a zero row: lanes with m>=4 read it
    // unconditionally instead of exec-masked predicated loads.
    __shared__ float s_a[(BM + 1) * BZ];    // W[m,z]*inv_nrm[z] + zero row (20 KB)
    __shared__ float s_inv[BZ];             // 1/max(||row||,eps)           (4 KB)
    __shared__ float s_g[(BM + 1) * GSTR];  // g + zero row                 (16 KB)
    __shared__ float s_div[256];
    __shared__ float s_cw[BM * BW];
    __shared__ float s_ch[BM * BH];
    __shared__ float s_gw[BM * BW];
    __shared__ float s_gh[BM * BH];
    __shared__ float s_den[BM];

    const float* xb = x + (size_t)b * BZ * BN;

    // ---------------- Phase 0: trig tables + zero pad rows ----------------
    s_div[tid & 255] = __expf((float)(tid & 255) * DIV_COEF);
    for (int z = tid; z < BZ; z += 256) s_a[BM * BZ + z] = 0.f;        // zero row
    for (int n = tid; n < GSTR; n += 256) s_g[BM * GSTR + n] = 0.f;    // zero row
    __syncthreads();

    if (tid < 224) {
        const bool  isW = (tid < 112);
        const int   t2  = isW ? tid : (tid - 112);
        const int   m   = t2 / 28;
        const int   p   = t2 % 28;
        const float* wrow = Wm + m * BZ + (isW ? 0 : DHALF);
        float acc = 0.f;
        for (int k = 0; k < 256; ++k) {
            float s, c;
            __sincosf((float)p * s_div[k], &s, &c);
            acc = fmaf(wrow[2 * k], s, acc);
            acc = fmaf(wrow[2 * k + 1], c, acc);
        }
        if (isW) s_cw[t2] = acc; else s_ch[t2] = acc;
    }

    // ---------------- Phase 1: row norms (coalesced, wave per row) --------
    for (int z = wave; z < BZ; z += 8) {
        const float* row = xb + (size_t)z * BN;
        float ss = 0.f;
        for (int n = lane; n < BN; n += 32) { float v = row[n]; ss = fmaf(v, v, ss); }
        #pragma unroll
        for (int off = 16; off > 0; off >>= 1) ss += __shfl_xor(ss, off, 32);
        if (lane == 0) s_inv[z] = 1.0f / fmaxf(sqrtf(ss), 1e-10f);
    }
    __syncthreads();

    for (int z = tid; z < BZ; z += 256) {
        const float inv = s_inv[z];
        #pragma unroll
        for (int m = 0; m < BM; ++m) s_a[m * BZ + z] = Wm[m * BZ + z] * inv;
    }
    __syncthreads();

    const int li = lane & 15;               // A: M row / B: N col
    const int kh = lane >> 4;               // K half-select (K = 2*kh, 2*kh+1)
    const int r  = (li < BM) ? li : BM;     // clamp to zero row for m>=4

    // ---------------- Phase 2: t = a(16x1024) @ x(1024x784), g=sigmoid ----
    {
        const float* ap = s_a + r * BZ + 2 * kh;           // ds_load_b64 source
        for (int tile = wave; tile < 49; tile += 8) {
            const int n0 = tile * 16;
            const float* bp = xb + (size_t)(2 * kh) * BN + n0 + li;
            v8f c = {};
            for (int z = 0; z < BZ; z += 16) {
                if ((z & 255) == 0 && z + 256 < BZ)
                    __builtin_prefetch(bp + (size_t)(z + 256) * BN, 0, 1);
                const v2f a0 = *(const v2f*)(ap + z);
                const v2f a1 = *(const v2f*)(ap + z + 4);
                const v2f a2 = *(const v2f*)(ap + z + 8);
                const v2f a3 = *(const v2f*)(ap + z + 12);
                const float* bz = bp + (size_t)z * BN;
                v2f b0, b1, b2, b3;
                b0.x = bz[0 * BN];  b0.y = bz[1 * BN];
                b1.x = bz[4 * BN];  b1.y = bz[5 * BN];
                b2.x = bz[8 * BN];  b2.y = bz[9 * BN];
                b3.x = bz[12 * BN]; b3.y = bz[13 * BN];
                c = __builtin_amdgcn_wmma_f32_16x16x4_f32(false, a0, false, b0, (short)0, c, false, false);
                c = __builtin_amdgcn_wmma_f32_16x16x4_f32(false, a1, false, b1, (short)0, c, false, false);
                c = __builtin_amdgcn_wmma_f32_16x16x4_f32(false, a2, false, b2, (short)0, c, false, false);
                c = __builtin_amdgcn_wmma_f32_16x16x4_f32(false, a3, false, b3, (short)0, c, false, false);
            }
            // D layout: lanes 0..15 hold rows M = vgpr index (0..7); need m=0..3
            if (lane < 16) {
                const int n = n0 + lane;
                const int h = n / BW, w = n % BW;
                #pragma unroll
                for (int m = 0; m < BM; ++m) {
                    const float t = c[m] + s_cw[m * BW + w] + s_ch[m * BH + h];
                    s_g[m * GSTR + n] = 1.0f / (1.0f + __expf(-t));
                }
            }
        }
    }
    __syncthreads();

    // ---------------- Phase 2.5: den, gw, gh reductions -------------------
    if (wave < BM) {
        float d = 0.f;
        for (int n = lane; n < BN; n += 32) { float gv = s_g[wave * GSTR + n]; d = fmaf(gv, gv, d); }
        #pragma unroll
        for (int off = 16; off > 0; off >>= 1) d += __shfl_xor(d, off, 32);
        if (lane == 0) s_den[wave] = d;
    }
    if (tid < 112) {                                  // gw[m,w] = sum_h g
        const int m = tid / 28, w = tid % 28;
        float acc = 0.f;
        for (int h = 0; h < BH; ++h) acc += s_g[m * GSTR + h * BW + w];
        s_gw[tid] = acc;
    } else if (tid < 224) {                           // gh[m,h] = sum_w g
        const int t2 = tid - 112, m = t2 / 28, h = t2 % 28;
        float acc = 0.f;
        for (int w = 0; w < BW; ++w) acc += s_g[m * GSTR + h * BW + w];
        s_gh[t2] = acc;
    }
    __syncthreads();

    // ---------------- Phase 3: S = g(16x784) @ x^T(784x1024), combine -----
    {
        const float* gp = s_g + r * GSTR + 2 * kh;          // ds_load_b64 source
        for (int tile = wave; tile < 64; tile += 8) {
            const int z0 = tile * 16;
            // B[k=n][col=z]=x[z,n]: per-lane row stream, consecutive pair -> b64.
            const float* rp = xb + (size_t)(z0 + li) * BN + 2 * kh;
            v8f c = {};
            for (int n = 0; n < BN; n += 16) {
                const v2f a0 = *(const v2f*)(gp + n);
                const v2f a1 = *(const v2f*)(gp + n + 4);
                const v2f a2 = *(const v2f*)(gp + n + 8);
                const v2f a3 = *(const v2f*)(gp + n + 12);
                // last use of x: non-temporal to keep L2 for other batches
                const v2f b0 = __builtin_nontemporal_load((const v2f*)(rp + n));
                const v2f b1 = __builtin_nontemporal_load((const v2f*)(rp + n + 4));
                const v2f b2 = __builtin_nontemporal_load((const v2f*)(rp + n + 8));
                const v2f b3 = __builtin_nontemporal_load((const v2f*)(rp + n + 12));
                c = __builtin_amdgcn_wmma_f32_16x16x4_f32(false, a0, false, b0, (short)0, c, false, false);
                c = __builtin_amdgcn_wmma_f32_16x16x4_f32(false, a1, false, b1, (short)0, c, false, false);
                c = __builtin_amdgcn_wmma_f32_16x16x4_f32(false, a2, false, b2, (short)0, c, false, false);
                c = __builtin_amdgcn_wmma_f32_16x16x4_f32(false, a3, false, b3, (short)0, c, false, false);
            }
            if (lane < 16) {
                const int z  = z0 + lane;
                const int zz = (z < DHALF) ? z : (z - DHALF);
                const float dv = s_div[zz >> 1];
                const bool useCos = (z & 1);
                const float* tab = (z < DHALF) ? s_gw : s_gh;
                float pev0 = 0.f, pev1 = 0.f, pev2 = 0.f, pev3 = 0.f;
                for (int p = 0; p < 28; ++p) {
                    float s, ct;
                    __sincosf((float)p * dv, &s, &ct);
                    const float tr = useCos ? ct : s;
                    pev0 = fmaf(tr, tab[0 * 28 + p], pev0);
                    pev1 = fmaf(tr, tab[1 * 28 + p], pev1);
                    pev2 = fmaf(tr, tab[2 * 28 + p], pev2);
                    pev3 = fmaf(tr, tab[3 * 28 + p], pev3);
                }
                const float inv = s_inv[z];
                float o = 0.f;
                o += (fmaf(inv, c[0], pev0)) / s_den[0];
                o += (fmaf(inv, c[1], pev1)) / s_den[1];
                o += (fmaf(inv, c[2], pev2)) / s_den[2];
                o += (fmaf(inv, c[3], pev3)) / s_den[3];
                out[(size_t)b * BZ + z] = o;
            }
        }
    }
}

extern "C" void kernel_launch(void* const* d_in, const int* in_sizes, int n_in,
                              void* d_out, int out_size, void* d_ws, size_t ws_size,
                              hipStream_t stream) {
    (void)in_sizes; (void)n_in; (void)d_ws; (void)ws_size; (void)out_size;
    const float* x  = (const float*)d_in[0];   // [128, 1024, 28, 28] fp32
    const float* Wm = (const float*)d_in[1];   // [4, 1024] fp32 (row-normalized)
    float* out = (float*)d_out;                // [128, 1, 1024] fp32
    rae_fused_kernel<<<128, 256, 0, stream>>>(x, Wm, out);
}